// CustomGINLayer_55027120996496
// MI455X (gfx1250) — compile-verified
//
#include <hip/hip_runtime.h>

#define N_NODES 50000
#define N_EDGES 600000
#define EMB     128
#define HID     256   // 2*EMB

typedef __attribute__((ext_vector_type(2))) float v2f;
typedef __attribute__((ext_vector_type(4))) float v4f;
typedef __attribute__((ext_vector_type(8))) float v8f;

// ---------------------------------------------------------------------------
// Kernel 1: zero the aggregation buffer (atomics accumulate into it)
// ---------------------------------------------------------------------------
__global__ void gin_zero(float* __restrict__ p, int n4) {
    int i = blockIdx.x * blockDim.x + threadIdx.x;
    if (i < n4) {
        v4f z = {0.f, 0.f, 0.f, 0.f};
        reinterpret_cast<v4f*>(p)[i] = z;
    }
}

// ---------------------------------------------------------------------------
// Kernel 2: per-edge gather + scatter-add.
// One wave32 per edge; each lane handles 4 contiguous floats (32*4 = 128).
// aggr (25.6 MB) fits in L2, so the f32 atomics resolve in L2.
// ---------------------------------------------------------------------------
__global__ void gin_edges(const float* __restrict__ h,
                          const float* __restrict__ edge_attr,
                          const int*   __restrict__ src,
                          const int*   __restrict__ dst,
                          float*       __restrict__ aggr) {
    int wave = (blockIdx.x * blockDim.x + threadIdx.x) >> 5;
    int lane = threadIdx.x & 31;
    if (wave >= N_EDGES) return;

    int s = src[wave];
    int d = dst[wave];

    const v4f hv = *reinterpret_cast<const v4f*>(h         + (size_t)s    * EMB + lane * 4);
    const v4f ev = *reinterpret_cast<const v4f*>(edge_attr + (size_t)wave * EMB + lane * 4);
    v4f m = hv + ev;

    float* out = aggr + (size_t)d * EMB + lane * 4;
    unsafeAtomicAdd(out + 0, m[0]);   // -> global_atomic_add_f32 (no return)
    unsafeAtomicAdd(out + 1, m[1]);
    unsafeAtomicAdd(out + 2, m[2]);
    unsafeAtomicAdd(out + 3, m[3]);
}

// ---------------------------------------------------------------------------
// WMMA f32 16x16x4 register layouts (ISA 7.12.2, wave32):
//   A (16x4, MxK): lane -> M = lane%16; VGPR v -> K = 2*(lane/16) + v
//   B (4x16, KxN): lane -> N = lane%16; VGPR v -> K = 2*(lane/16) + v
//   C/D (16x16) :  lane -> N = lane%16; VGPR v -> M = v + 8*(lane/16)
// ---------------------------------------------------------------------------

// Kernel 3: hidden = relu(aggr @ W1 + b1)   [50000x128] x [128x256]
__global__ void gin_gemm1(const float* __restrict__ aggr,
                          const float* __restrict__ W1,
                          const float* __restrict__ b1,
                          float*       __restrict__ hidden) {
    const int nTiles = HID / 16;                       // 16
    int wave = (blockIdx.x * blockDim.x + threadIdx.x) >> 5;
    int lane = threadIdx.x & 31;
    if (wave >= (N_NODES / 16) * nTiles) return;

    int tm = wave / nTiles;
    int tn = wave % nTiles;
    int l16   = lane & 15;
    int kGrp  = lane >> 4;                             // 0 or 1
    int row_m = tm * 16 + l16;                         // A row this lane loads
    int col_n = tn * 16 + l16;                         // B col this lane loads

    v8f acc = {};
    const float* arow = aggr + (size_t)row_m * EMB + 2 * kGrp;
#pragma unroll 4
    for (int kb = 0; kb < EMB; kb += 4) {
        v2f a = *reinterpret_cast<const v2f*>(arow + kb);
        v2f b;
        b[0] = W1[(size_t)(kb + 2 * kGrp + 0) * HID + col_n];
        b[1] = W1[(size_t)(kb + 2 * kGrp + 1) * HID + col_n];
        acc = __builtin_amdgcn_wmma_f32_16x16x4_f32(
                  false, a, false, b, (short)0, acc, false, false);
    }

    float bias = b1[col_n];
#pragma unroll
    for (int v = 0; v < 8; ++v) {
        int m = tm * 16 + v + 8 * kGrp;
        float val = acc[v] + bias;
        hidden[(size_t)m * HID + col_n] = fmaxf(val, 0.f);
    }
}

// Kernel 4: out = (1+eps)*h + hidden @ W2 + b2   [50000x256] x [256x128]
__global__ void gin_gemm2(const float* __restrict__ hidden,
                          const float* __restrict__ W2,
                          const float* __restrict__ b2,
                          const float* __restrict__ h,
                          const float* __restrict__ eps,
                          float*       __restrict__ out) {
    const int nTiles = EMB / 16;                       // 8
    int wave = (blockIdx.x * blockDim.x + threadIdx.x) >> 5;
    int lane = threadIdx.x & 31;
    if (wave >= (N_NODES / 16) * nTiles) return;

    int tm = wave / nTiles;
    int tn = wave % nTiles;
    int l16   = lane & 15;
    int kGrp  = lane >> 4;
    int row_m = tm * 16 + l16;
    int col_n = tn * 16 + l16;

    v8f acc = {};
    const float* arow = hidden + (size_t)row_m * HID + 2 * kGrp;
#pragma unroll 4
    for (int kb = 0; kb < HID; kb += 4) {
        v2f a = *reinterpret_cast<const v2f*>(arow + kb);
        v2f b;
        b[0] = W2[(size_t)(kb + 2 * kGrp + 0) * EMB + col_n];
        b[1] = W2[(size_t)(kb + 2 * kGrp + 1) * EMB + col_n];
        acc = __builtin_amdgcn_wmma_f32_16x16x4_f32(
                  false, a, false, b, (short)0, acc, false, false);
    }

    float bias  = b2[col_n];
    float scale = 1.0f + eps[0];
#pragma unroll
    for (int v = 0; v < 8; ++v) {
        int m = tm * 16 + v + 8 * kGrp;
        float val = acc[v] + bias;
        out[(size_t)m * EMB + col_n] = scale * h[(size_t)m * EMB + col_n] + val;
    }
}

// ---------------------------------------------------------------------------
extern "C" void kernel_launch(void* const* d_in, const int* in_sizes, int n_in,
                              void* d_out, int out_size, void* d_ws, size_t ws_size,
                              hipStream_t stream) {
    const float* h         = (const float*)d_in[0];
    const float* edge_attr = (const float*)d_in[1];
    const int*   src       = (const int*)  d_in[2];
    const int*   dst       = (const int*)  d_in[3];
    const float* W1        = (const float*)d_in[4];
    const float* b1        = (const float*)d_in[5];
    const float* W2        = (const float*)d_in[6];
    const float* b2        = (const float*)d_in[7];
    const float* eps       = (const float*)d_in[8];
    float*       out       = (float*)d_out;

    float* aggr   = (float*)d_ws;                        // 50000*128 f32 = 25.6 MB
    float* hidden = aggr + (size_t)N_NODES * EMB;        // 50000*256 f32 = 51.2 MB

    // 1) zero aggr
    {
        int n4 = N_NODES * EMB / 4;
        gin_zero<<<(n4 + 255) / 256, 256, 0, stream>>>(aggr, n4);
    }
    // 2) edge gather + scatter-add (1 wave per edge, 8 waves/block)
    gin_edges<<<(N_EDGES + 7) / 8, 256, 0, stream>>>(h, edge_attr, src, dst, aggr);

    // 3) GEMM1 + bias + relu   (50000/16 * 256/16 = 50000 waves)
    {
        int waves = (N_NODES / 16) * (HID / 16);
        gin_gemm1<<<(waves + 7) / 8, 256, 0, stream>>>(aggr, W1, b1, hidden);
    }
    // 4) GEMM2 + bias + (1+eps)*h  (50000/16 * 128/16 = 25000 waves)
    {
        int waves = (N_NODES / 16) * (EMB / 16);
        gin_gemm2<<<(waves + 7) / 8, 256, 0, stream>>>(hidden, W2, b2, h, eps, out);
    }
}